// GNN_GCNLSTM_48653389529157
// MI455X (gfx1250) — compile-verified
//
#include <hip/hip_runtime.h>
#include <hip/hip_bf16.h>
#include <math.h>

#define N_NODES 30000
#define N_EDGES 480000
#define T_STEPS 8
#define F_IN 8
#define HID 32
#define KCH 5
#define PERIODS 8
#define BC 200   /* basis columns: 5*8 (x terms) + 5*32 (H terms) */
#define GC 128   /* gate columns: 4 gates * 32 hidden */

typedef __attribute__((ext_vector_type(2))) float v2f;
typedef __attribute__((ext_vector_type(8))) float v8f;

__device__ __forceinline__ void atomAddF(float* p, float v) {
    unsafeAtomicAdd(p, v);  // lowers to global_atomic_add_f32 (L2 atomics)
}

// ---------------- setup kernels ----------------

__global__ void k_zero(float* __restrict__ p, int n) {
    int i = blockIdx.x * blockDim.x + threadIdx.x;
    if (i < n) p[i] = 0.f;
}

__global__ void k_deg(const int* __restrict__ src, const int* __restrict__ dst,
                      float* __restrict__ deg) {
    int e = blockIdx.x * blockDim.x + threadIdx.x;
    if (e < N_EDGES) {
        int s = src[e];
        if (s != dst[e]) atomAddF(&deg[s], 1.0f);
    }
}

__global__ void k_norm(const int* __restrict__ src, const int* __restrict__ dst,
                       const float* __restrict__ deg, float* __restrict__ norm) {
    int e = blockIdx.x * blockDim.x + threadIdx.x;
    if (e >= N_EDGES) return;
    int s = src[e], d = dst[e];
    float v = 0.f;
    if (s != d) {
        float ds = deg[s], dd = deg[d];
        float is = ds > 0.f ? rsqrtf(ds) : 0.f;
        float id = dd > 0.f ? rsqrtf(dd) : 0.f;
        v = is * id;
    }
    norm[e] = v;
}

// Pack Wx[4,5,8,32] and Wh[4,5,32,32] into row-pair-interleaved Wp:
// logical Wcat[row=0..199][col=0..127] with
//   row = k*8+f (x part) or 40 + k*32 + i (H part); col = g*32 + h.
// stored as Wp[(row/2)*256 + col*2 + (row&1)] so a lane fetches
// {Wcat[kb][col], Wcat[kb+1][col]} with ONE global_load_b64.
__global__ void k_pack(const float* __restrict__ Wx, const float* __restrict__ Wh,
                       float* __restrict__ Wp) {
    int i = blockIdx.x * blockDim.x + threadIdx.x;
    if (i >= BC * GC) return;
    int row = i >> 7, col = i & 127;
    int g = col >> 5, h = col & 31;
    float v;
    if (row < KCH * F_IN) {
        int k = row >> 3, f = row & 7;
        v = Wx[((g * KCH + k) * F_IN + f) * HID + h];
    } else {
        int rr = row - KCH * F_IN;
        int k = rr >> 5, ii = rr & 31;
        v = Wh[((g * KCH + k) * HID + ii) * HID + h];
    }
    Wp[(row >> 1) * (2 * GC) + col * 2 + (row & 1)] = v;
}

// ---------------- per-timestep kernels ----------------

// Seed Chebyshev term 0: basis[:,0:8] = x_t ; basis[:,40:72] = H
__global__ void k_seed(const float* __restrict__ x, const float* __restrict__ H,
                       float* __restrict__ basis) {
    int i = blockIdx.x * blockDim.x + threadIdx.x;
    if (i >= N_NODES * HID) return;
    int n = i >> 5, h = i & 31;
    size_t row = (size_t)n * BC;
    basis[row + KCH * F_IN + h] = H[i];
    if (h < F_IN) basis[row + h] = x[n * F_IN + h];
}

// Initialize term k: 0 (k==1) or -T_{k-2} (k>=2); scatter adds the Laplacian part.
__global__ void k_lap_init(float* __restrict__ basis, int colOut, int colPP, int C, int total) {
    int i = blockIdx.x * blockDim.x + threadIdx.x;
    if (i >= total) return;
    int n = i / C, c = i - n * C;
    size_t row = (size_t)n * BC;
    basis[row + colOut + c] = (colPP >= 0) ? -basis[row + colPP + c] : 0.f;
}

// T_k[dst] += -coef * norm[e] * T_{k-1}[src]   (lap(x) = -segsum(norm*x[src]))
template <int C>
__global__ void k_scatter(const int* __restrict__ src, const int* __restrict__ dst,
                          const float* __restrict__ norm, float* __restrict__ basis,
                          int colIn, int colOut, float coef) {
    int e = blockIdx.x * blockDim.x + threadIdx.x;
    if (e >= N_EDGES) return;
    float nm = norm[e];
    if (nm == 0.f) return;
    float scale = -coef * nm;
    const float* in = basis + (size_t)src[e] * BC + colIn;
    float* out = basis + (size_t)dst[e] * BC + colOut;
#pragma unroll
    for (int c = 0; c < C; ++c) atomAddF(&out[c], scale * in[c]);
}

// gates[N,128] = basis[N,200] @ Wcat[200,128] via V_WMMA_F32_16X16X4_F32.
// One block = 8 waves = one 16-row tile (8 col tiles of 16). EXEC all-ones.
__global__ void __launch_bounds__(256) k_gemm(const float* __restrict__ basis,
                                              const float* __restrict__ Wp,
                                              float* __restrict__ gates) {
    int lane = threadIdx.x & 31;
    int tc = threadIdx.x >> 5;     // col tile 0..7
    int tr = blockIdx.x;           // row tile 0..1874
    int r0 = tr << 4, c0 = tc << 4;
    int m = lane & 15;             // M row (A) / N col (B,C,D)
    int hi = lane >> 4;            // 0 or 1 (K sub-group)
    int kb = hi << 1;              // K sub-offset: 0 or 2
    const float* aptr = basis + (size_t)(r0 + m) * BC + kb;       // float2-aligned
    // pair index for this lane at k0: (k0+kb)/2 = k0/2 + hi
    const float* bptr = Wp + (size_t)hi * (2 * GC) + (size_t)(c0 + m) * 2;
    v8f acc = {};
#pragma unroll 5
    for (int k0 = 0; k0 < BC; k0 += 4) {
        v2f a = *(const v2f*)(aptr + k0);                         // A[m][k0+kb .. +1]
        v2f b = *(const v2f*)(bptr + (size_t)(k0 >> 1) * (2 * GC)); // B[k0+kb .. +1][c0+m]
        acc = __builtin_amdgcn_wmma_f32_16x16x4_f32(
            /*neg_a=*/false, a, /*neg_b=*/false, b,
            /*c_mod=*/(short)0, acc, /*reuse_a=*/false, /*reuse_b=*/false);
    }
    int rowBase = r0 + (hi << 3);                                 // M = r + 8*(lane/16)
    float* gout = gates + (size_t)rowBase * GC + c0 + m;
#pragma unroll
    for (int r = 0; r < 8; ++r) gout[(size_t)r * GC] = acc[r];
}

// LSTM pointwise update with peepholes.
__global__ void k_lstm(const float* __restrict__ gates, const float* __restrict__ w_c,
                       const float* __restrict__ b_g, float* __restrict__ H,
                       float* __restrict__ C) {
    int i = blockIdx.x * blockDim.x + threadIdx.x;
    if (i >= N_NODES * HID) return;
    int n = i >> 5, h = i & 31;
    const float* g = gates + (size_t)n * GC;
    float c_old = C[i];
    float gi = g[h], gf = g[32 + h], gc = g[64 + h], go = g[96 + h];
    float I = 1.f / (1.f + expf(-(gi + w_c[h] * c_old + b_g[h])));
    float F = 1.f / (1.f + expf(-(gf + w_c[32 + h] * c_old + b_g[32 + h])));
    float T = tanhf(gc + b_g[64 + h]);
    float Cn = F * c_old + I * T;
    float O = 1.f / (1.f + expf(-(go + w_c[64 + h] * Cn + b_g[96 + h])));
    H[i] = O * tanhf(Cn);
    C[i] = Cn;
}

// out[n,p] = relu(H[n,:]) . W_lin[p,:] + b_lin[p]
__global__ void k_out(const float* __restrict__ H, const float* __restrict__ Wl,
                      const float* __restrict__ bl, float* __restrict__ out) {
    int i = blockIdx.x * blockDim.x + threadIdx.x;
    if (i >= N_NODES * PERIODS) return;
    int n = i >> 3, p = i & 7;
    const float* hrow = H + (size_t)n * HID;
    const float* wrow = Wl + (size_t)p * HID;
    float acc = bl[p];
#pragma unroll
    for (int h = 0; h < HID; ++h) acc += fmaxf(hrow[h], 0.f) * wrow[h];
    out[i] = acc;
}

// ---------------- host driver ----------------

extern "C" void kernel_launch(void* const* d_in, const int* in_sizes, int n_in,
                              void* d_out, int out_size, void* d_ws, size_t ws_size,
                              hipStream_t stream) {
    const float* timesteps = (const float*)d_in[0];
    const int*   edge      = (const int*)d_in[1];
    const int*   src       = edge;
    const int*   dst       = edge + N_EDGES;
    const float* Wx        = (const float*)d_in[2];
    const float* Wh        = (const float*)d_in[3];
    const float* w_c       = (const float*)d_in[4];
    const float* b_g       = (const float*)d_in[5];
    const float* Wl        = (const float*)d_in[6];
    const float* bl        = (const float*)d_in[7];
    float* out = (float*)d_out;

    float* ws    = (float*)d_ws;
    float* norm  = ws;  ws += N_EDGES;
    float* deg   = ws;  ws += N_NODES;
    float* Hst   = ws;  ws += (size_t)N_NODES * HID;
    float* Cst   = ws;  ws += (size_t)N_NODES * HID;  // contiguous after Hst
    float* basis = ws;  ws += (size_t)N_NODES * BC;
    float* gates = ws;  ws += (size_t)N_NODES * GC;
    float* Wp    = ws;  ws += BC * GC;

    const int TB = 256;
    auto blocks = [](int n) { return (n + 255) / 256; };

    k_zero<<<blocks(N_NODES), TB, 0, stream>>>(deg, N_NODES);
    k_zero<<<blocks(2 * N_NODES * HID), TB, 0, stream>>>(Hst, 2 * N_NODES * HID); // H and C
    k_deg<<<blocks(N_EDGES), TB, 0, stream>>>(src, dst, deg);
    k_norm<<<blocks(N_EDGES), TB, 0, stream>>>(src, dst, deg, norm);
    k_pack<<<blocks(BC * GC), TB, 0, stream>>>(Wx, Wh, Wp);

    for (int t = 0; t < T_STEPS; ++t) {
        const float* x = timesteps + (size_t)t * N_NODES * F_IN;
        k_seed<<<blocks(N_NODES * HID), TB, 0, stream>>>(x, Hst, basis);

        // Chebyshev recurrence on x part (cols 0..39, C=8)
        for (int k = 1; k < KCH; ++k) {
            int colOut = k * F_IN, colIn = (k - 1) * F_IN;
            int colPP = (k >= 2) ? (k - 2) * F_IN : -1;
            float coef = (k >= 2) ? 2.f : 1.f;
            k_lap_init<<<blocks(N_NODES * F_IN), TB, 0, stream>>>(basis, colOut, colPP, F_IN,
                                                                  N_NODES * F_IN);
            k_scatter<F_IN><<<blocks(N_EDGES), TB, 0, stream>>>(src, dst, norm, basis,
                                                                colIn, colOut, coef);
        }
        // Chebyshev recurrence on H part (cols 40..199, C=32)
        for (int k = 1; k < KCH; ++k) {
            int base = KCH * F_IN;
            int colOut = base + k * HID, colIn = base + (k - 1) * HID;
            int colPP = (k >= 2) ? base + (k - 2) * HID : -1;
            float coef = (k >= 2) ? 2.f : 1.f;
            k_lap_init<<<blocks(N_NODES * HID), TB, 0, stream>>>(basis, colOut, colPP, HID,
                                                                 N_NODES * HID);
            k_scatter<HID><<<blocks(N_EDGES), TB, 0, stream>>>(src, dst, norm, basis,
                                                               colIn, colOut, coef);
        }

        k_gemm<<<N_NODES / 16, 256, 0, stream>>>(basis, Wp, gates);
        k_lstm<<<blocks(N_NODES * HID), TB, 0, stream>>>(gates, w_c, b_g, Hst, Cst);
    }
    k_out<<<blocks(N_NODES * PERIODS), TB, 0, stream>>>(Hst, Wl, bl, out);
}